// SimpleGCN_8899172237583
// MI455X (gfx1250) — compile-verified
//
#include <hip/hip_runtime.h>

// ---------------------------------------------------------------------------
// SimpleGCN forward for MI455X (gfx1250, wave32).
// GEMMs use V_WMMA_F32_16X16X4_F32 (full fp32 precision, 16x16 C-tile/wave).
// Edge aggregation uses global_atomic_add_f32 resolved in the 192MB L2
// (feature matrices are 51.2MB each -> L2 resident working set).
// ---------------------------------------------------------------------------

typedef __attribute__((ext_vector_type(2))) float v2f;
typedef __attribute__((ext_vector_type(8))) float v8f;

#define DCH 128   // feature dim (D_IN == D_HID == 128)

// ---------------- small elementwise helpers ----------------

__global__ void fill_f32(float* __restrict__ p, float v, int n) {
  int i = blockIdx.x * blockDim.x + threadIdx.x;
  if (i < n) p[i] = v;
}

__global__ void deg_count(const int* __restrict__ dst, float* __restrict__ deg, int nE) {
  int i = blockIdx.x * blockDim.x + threadIdx.x;
  if (i < nE) atomicAdd(&deg[dst[i]], 1.0f);
}

__global__ void deg_to_dinv(float* __restrict__ deg, int n) {
  int i = blockIdx.x * blockDim.x + threadIdx.x;
  if (i < n) {
    float d = deg[i];
    deg[i] = d > 0.0f ? rsqrtf(d) : 0.0f;   // deg >= 1 (self loop), but keep ref semantics
  }
}

// out[i,c] = h[i,c] * dinv[i]^2   (self-loop contribution initializes accumulator)
__global__ void self_loop_init(const float* __restrict__ h, const float* __restrict__ dinv,
                               float* __restrict__ out, int nElems) {
  int i = blockIdx.x * blockDim.x + threadIdx.x;
  if (i < nElems) {
    float di = dinv[i >> 7];
    out[i] = h[i] * di * di;
  }
}

// ---------------- WMMA fp32 GEMM: H[M,128] = op(A)[M,128] @ W[128,128] ----------------
// One wave computes one 16x16 output tile; K advanced 4 at a time via
// v_wmma_f32_16x16x4_f32. FUSE_RELU_BIAS applies relu(a + bias[k]) to the
// A operand on load (layer-2 input activation fused).

template <bool FUSE_RELU_BIAS>
__global__ void gemm_wmma_f32(const float* __restrict__ A, const float* __restrict__ W,
                              const float* __restrict__ bias, float* __restrict__ H,
                              int M) {
  const int wave   = blockIdx.x * (blockDim.x >> 5) + (threadIdx.x >> 5);
  const int tile_m = wave >> 3;           // 128/16 = 8 N-tiles
  const int tile_n = wave & 7;
  if (tile_m * 16 >= M) return;

  const int lane = threadIdx.x & 31;
  const int half = lane >> 4;             // 0: lanes 0-15, 1: lanes 16-31
  const int l    = lane & 15;

  int mrow = tile_m * 16 + l;             // A row for this lane
  if (mrow >= M) mrow = M - 1;            // clamp (M=100000 is divisible by 16)
  const int ncol = tile_n * 16 + l;       // B column for this lane

  v8f acc = {};
  for (int k0 = 0; k0 < DCH; k0 += 4) {
    const int ka = k0 + 2 * half;         // lanes 0-15: K=k0,k0+1; lanes 16-31: K=k0+2,k0+3
    float a0 = A[mrow * DCH + ka];
    float a1 = A[mrow * DCH + ka + 1];
    if (FUSE_RELU_BIAS) {
      a0 = fmaxf(a0 + bias[ka], 0.0f);
      a1 = fmaxf(a1 + bias[ka + 1], 0.0f);
    }
    v2f a; a.x = a0; a.y = a1;
    v2f b; b.x = W[ka * DCH + ncol];
           b.y = W[(ka + 1) * DCH + ncol];
    // D = A(16x4) x B(4x16) + C ; full fp32
    acc = __builtin_amdgcn_wmma_f32_16x16x4_f32(false, a, false, b,
                                                (short)0, acc, false, false);
  }

  // C/D layout: VGPR v -> M = tile_m*16 + v + 8*half, N = tile_n*16 + l
  const int rbase = tile_m * 16 + 8 * half;
#pragma unroll
  for (int v = 0; v < 8; ++v) {
    const int r = rbase + v;
    if (r < M) H[r * DCH + tile_n * 16 + l] = acc[v];
  }
}

// ---------------- edge scatter: out[dst] += h[src] * dinv[src]*dinv[dst] ----------------
// One wave per edge; each lane handles 4 contiguous channels (float4 gather,
// 4 global_atomic_add_f32). Accumulator + h are L2-resident.

__global__ void agg_edges(const float* __restrict__ h, const int* __restrict__ src,
                          const int* __restrict__ dst, const float* __restrict__ dinv,
                          float* __restrict__ out, int nE) {
  const int gid = blockIdx.x * blockDim.x + threadIdx.x;
  const int e = gid >> 5;
  if (e >= nE) return;
  const int lane = threadIdx.x & 31;
  const int s = src[e];
  const int d = dst[e];
  const float w = dinv[s] * dinv[d];
  const float4 hv = *reinterpret_cast<const float4*>(h + s * DCH + lane * 4);
  float* o = out + d * DCH + lane * 4;
  atomicAdd(o + 0, hv.x * w);
  atomicAdd(o + 1, hv.y * w);
  atomicAdd(o + 2, hv.z * w);
  atomicAdd(o + 3, hv.w * w);
}

// ---------------- pooling ----------------

__global__ void pool_accum(const float* __restrict__ h2, const int* __restrict__ batch,
                           float* __restrict__ pool, int nElems) {
  int i = blockIdx.x * blockDim.x + threadIdx.x;
  if (i < nElems) {
    const int node = i >> 7;
    const int c = i & (DCH - 1);
    atomicAdd(&pool[batch[node] * DCH + c], h2[i]);
  }
}

__global__ void cnt_accum(const int* __restrict__ batch, float* __restrict__ cnt, int n) {
  int i = blockIdx.x * blockDim.x + threadIdx.x;
  if (i < n) atomicAdd(&cnt[batch[i]], 1.0f);
}

// out[g] = sum_c (pool[g,c]/max(cnt,1) + b2[c]) * Wm[c] + bm   (wave per graph)
__global__ void final_proj(const float* __restrict__ pool, const float* __restrict__ cnt,
                           const float* __restrict__ b2, const float* __restrict__ Wm,
                           const float* __restrict__ bm, float* __restrict__ out, int G) {
  const int g = (blockIdx.x * blockDim.x + threadIdx.x) >> 5;
  if (g >= G) return;
  const int lane = threadIdx.x & 31;
  const float inv = 1.0f / fmaxf(cnt[g], 1.0f);
  float p = 0.0f;
  for (int c = lane; c < DCH; c += 32)
    p += (pool[g * DCH + c] * inv + b2[c]) * Wm[c];
  for (int off = 16; off > 0; off >>= 1)
    p += __shfl_down(p, off, 32);
  if (lane == 0) out[g] = p + bm[0];
}

// ---------------- driver ----------------

extern "C" void kernel_launch(void* const* d_in, const int* in_sizes, int n_in,
                              void* d_out, int out_size, void* d_ws, size_t ws_size,
                              hipStream_t stream) {
  const float* x     = (const float*)d_in[0];
  const int*   ei    = (const int*)  d_in[1];   // [2, E]
  const int*   batch = (const int*)  d_in[2];   // [N]
  const float* W1    = (const float*)d_in[3];
  const float* b1    = (const float*)d_in[4];
  const float* W2    = (const float*)d_in[5];
  const float* b2    = (const float*)d_in[6];
  const float* Wm    = (const float*)d_in[7];
  const float* bm    = (const float*)d_in[8];
  float* out = (float*)d_out;

  const int N = in_sizes[0] / DCH;     // 100000
  const int E = in_sizes[1] / 2;       // 1600000
  const int G = out_size;              // 256

  const int* src = ei;
  const int* dst = ei + E;

  // workspace layout (all fp32): dinv[N] | bufA[N*128] | bufB[N*128] | pool[G*128] | cnt[G]
  float* dinv = (float*)d_ws;
  float* bufA = dinv + N;
  float* bufB = bufA + (size_t)N * DCH;
  float* pool = bufB + (size_t)N * DCH;
  float* cnt  = pool + (size_t)G * DCH;

  const int BLK = 256;
  auto nblk = [](long long n, int b) { return (int)((n + b - 1) / b); };

  // --- degree / normalization: deg starts at 1 (self loop), +1 per incoming edge ---
  fill_f32<<<nblk(N, BLK), BLK, 0, stream>>>(dinv, 1.0f, N);
  deg_count<<<nblk(E, BLK), BLK, 0, stream>>>(dst, dinv, E);
  deg_to_dinv<<<nblk(N, BLK), BLK, 0, stream>>>(dinv, N);

  // --- layer 1: h1 = x @ W1 (WMMA fp32) ---
  const int tilesM = (N + 15) / 16;
  const int gemmBlocks = (tilesM * 8 + 7) / 8;   // 8 waves/block, 8 N-tiles per M-tile row
  gemm_wmma_f32<false><<<gemmBlocks, 256, 0, stream>>>(x, W1, nullptr, bufA, N);

  // --- aggregate 1: bufB = D^-1/2 A_hat D^-1/2 h1 ---
  self_loop_init<<<nblk((long long)N * DCH, BLK), BLK, 0, stream>>>(bufA, dinv, bufB, N * DCH);
  agg_edges<<<nblk((long long)E * 32, BLK), BLK, 0, stream>>>(bufA, src, dst, dinv, bufB, E);

  // --- layer 2: h2 = relu(agg1 + b1) @ W2 (bias+ReLU fused into A load) ---
  gemm_wmma_f32<true><<<gemmBlocks, 256, 0, stream>>>(bufB, W2, b1, bufA, N);

  // --- aggregate 2: bufB = D^-1/2 A_hat D^-1/2 h2 ---
  self_loop_init<<<nblk((long long)N * DCH, BLK), BLK, 0, stream>>>(bufA, dinv, bufB, N * DCH);
  agg_edges<<<nblk((long long)E * 32, BLK), BLK, 0, stream>>>(bufA, src, dst, dinv, bufB, E);

  // --- global mean pool + projection (b2 folded into final projection) ---
  fill_f32<<<nblk(G * DCH + G, BLK), BLK, 0, stream>>>(pool, 0.0f, G * DCH + G); // pool+cnt contiguous
  pool_accum<<<nblk((long long)N * DCH, BLK), BLK, 0, stream>>>(bufB, batch, pool, N * DCH);
  cnt_accum<<<nblk(N, BLK), BLK, 0, stream>>>(batch, cnt, N);
  final_proj<<<nblk((long long)G * 32, BLK), BLK, 0, stream>>>(pool, cnt, b2, Wm, bm, out, G);
}